// SelfAttention_3959959847664
// MI455X (gfx1250) — compile-verified
//
#include <hip/hip_runtime.h>
#include <math.h>

// CDNA5 / gfx1250, wave32. Fused additive-attention pipeline.
// f32 WMMA 16x16x4; b128-only hot loop via U_w transpose + commuted-K trick;
// async global->LDS staging of per-batch broadcast vectors.

typedef __attribute__((ext_vector_type(2))) float v2f;
typedef __attribute__((ext_vector_type(4))) float v4f;
typedef __attribute__((ext_vector_type(8))) float v8f;
typedef __attribute__((ext_vector_type(4))) int   v4i;

#define UNITS 1024
#define BATCH 32
#define SEQ   2048

// ---------------------------------------------------------------------------
// Async global->LDS copy (ASYNCcnt path), with inline-asm fallback.
__device__ inline void async_copy_b128(const void* gsrc, void* lds_dst) {
#if __has_builtin(__builtin_amdgcn_global_load_async_to_lds_b128)
  __builtin_amdgcn_global_load_async_to_lds_b128(
      (__attribute__((address_space(1))) v4i*)gsrc,
      (__attribute__((address_space(3))) v4i*)lds_dst, 0, 0);
#else
  unsigned lds_off =
      (unsigned)(__UINTPTR_TYPE__)(__attribute__((address_space(3))) void*)lds_dst;
  asm volatile("global_load_async_to_lds_b128 %0, %1, off"
               :: "v"(lds_off), "v"(gsrc) : "memory");
#endif
}
__device__ inline void wait_async0() {
#if __has_builtin(__builtin_amdgcn_s_wait_asynccnt)
  __builtin_amdgcn_s_wait_asynccnt(0);
#else
  asm volatile("s_wait_asynccnt 0x0" ::: "memory");
#endif
}

// ---------------------------------------------------------------------------
// T0: Ut[n][k] = U_w[k][n]   (1024x1024), LDS-tiled transpose.
__global__ void transpose_kernel(const float* __restrict__ src,
                                 float* __restrict__ dst) {
  __shared__ float tile[32][33];
  const int x  = blockIdx.x * 32 + threadIdx.x;
  const int y0 = blockIdx.y * 32;
  #pragma unroll
  for (int j = threadIdx.y; j < 32; j += 8)
    tile[j][threadIdx.x] = src[(size_t)(y0 + j) * UNITS + x];
  __syncthreads();
  const int xo  = blockIdx.y * 32 + threadIdx.x;
  const int yo0 = blockIdx.x * 32;
  #pragma unroll
  for (int j = threadIdx.y; j < 32; j += 8)
    dst[(size_t)(yo0 + j) * UNITS + xo] = tile[threadIdx.x][j];
}

// ---------------------------------------------------------------------------
// K1: wsb[b,n] = (s_prev @ W_w)[b,n] + W_b[n] + U_b[n]     (32 x 1024)
__global__ void ws_gemm_kernel(const float* __restrict__ s_prev,
                               const float* __restrict__ W_w,
                               const float* __restrict__ W_b,
                               const float* __restrict__ U_b,
                               float* __restrict__ wsb) {
  const int wave = threadIdx.x >> 5;
  const int lane = threadIdx.x & 31;
  const int lo = lane & 15, hi = lane >> 4;
  const int row0 = (wave >> 2) * 16;
  const int col0 = blockIdx.x * 64 + (wave & 3) * 16;

  const float* Abase = s_prev + (size_t)(row0 + lo) * UNITS + 2 * hi;
  const float* Bbase = W_w + (size_t)(2 * hi) * UNITS + col0 + lo;

  v8f acc = {};
  #pragma unroll 4
  for (int k = 0; k < UNITS; k += 4) {
    v2f a;  a.x = Abase[k];                   a.y = Abase[k + 1];
    v2f bb; bb.x = Bbase[(size_t)k * UNITS];  bb.y = Bbase[(size_t)(k + 1) * UNITS];
    acc = __builtin_amdgcn_wmma_f32_16x16x4_f32(false, a, false, bb,
                                                (short)0, acc, false, false);
  }
  const int n = col0 + lo;
  const float bias = W_b[n] + U_b[n];
  #pragma unroll
  for (int r = 0; r < 8; ++r)
    wsb[(size_t)(row0 + r + 8 * hi) * UNITS + n] = acc[r] + bias;
}

// ---------------------------------------------------------------------------
// K2: score[b,s] = sum_n tanh(wsb[b,n] + (hidden[b] @ U_w)[s,n]) * V_w[n] + V_b
// One WG per (b, s-tile of 32); 8 waves tile 32(s) x 64(n); loop 16 n-chunks.
// Hot loop: per 8 K-values -> 1 A b128 + 1 Bt b128 + 2 WMMAs (commuted-K).
__global__ void score_kernel(const float* __restrict__ hidden,  // [B,S,U]
                             const float* __restrict__ Ut,      // [U,U] transposed
                             const float* __restrict__ V_w,     // [U]
                             const float* __restrict__ V_b,     // [1]
                             const float* __restrict__ wsb,     // [B,U]
                             float* __restrict__ score) {       // [B,S]
  __shared__ __align__(16) float s_wsb[UNITS];
  __shared__ __align__(16) float s_vw[UNITS];
  __shared__ float s_score[32];

  const int b  = blockIdx.x >> 6;
  const int s0 = (blockIdx.x & 63) * 32;
  const int wave = threadIdx.x >> 5;
  const int lane = threadIdx.x & 31;
  const int lo = lane & 15, hi = lane >> 4;
  const int sT = wave >> 2;   // 0..1 : s sub-tile
  const int nT = wave & 3;    // 0..3 : n sub-tile within 64-wide chunk

  // Stage broadcast vectors into LDS via async global->LDS (ASYNCcnt).
  {
    const int t4 = threadIdx.x * 4;
    async_copy_b128(wsb + (size_t)b * UNITS + t4, &s_wsb[t4]);
    async_copy_b128(V_w + t4, &s_vw[t4]);
  }
  if (threadIdx.x < 32) s_score[threadIdx.x] = 0.0f;
  wait_async0();
  __syncthreads();

  // Per-lane b128 covers K{k+4hi..k+4hi+3}; lanes 0-15 / 16-31 split K
  // consistently between A and Bt, so each WMMA contracts a valid K subset.
  const float* Abase =
      hidden + ((size_t)b * SEQ + s0 + sT * 16 + lo) * UNITS + 4 * hi;

  float pscore[8];
  #pragma unroll
  for (int r = 0; r < 8; ++r) pscore[r] = 0.0f;

  for (int n0 = 0; n0 < UNITS; n0 += 64) {
    const int col = n0 + nT * 16 + lo;
    const float* Bbase = Ut + (size_t)col * UNITS + 4 * hi;
    if (n0 + 64 < UNITS)   // prefetch next Bt panel -> global_prefetch_b8
      __builtin_prefetch(Bbase + (size_t)64 * UNITS, 0, 1);

    v8f acc = {};
    #pragma unroll 4
    for (int k = 0; k < UNITS; k += 8) {
      const v4f a  = *(const v4f*)(Abase + k);
      const v4f bt = *(const v4f*)(Bbase + k);
      const v2f a01 = {a.x, a.y},  a23 = {a.z, a.w};
      const v2f b01 = {bt.x, bt.y}, b23 = {bt.z, bt.w};
      acc = __builtin_amdgcn_wmma_f32_16x16x4_f32(false, a01, false, b01,
                                                  (short)0, acc, false, false);
      acc = __builtin_amdgcn_wmma_f32_16x16x4_f32(false, a23, false, b23,
                                                  (short)0, acc, false, false);
    }

    const float wsn = s_wsb[col];
    const float vw  = s_vw[col];
    #pragma unroll
    for (int r = 0; r < 8; ++r)
      pscore[r] += tanhf(wsn + acc[r]) * vw;
  }

  // One LDS-atomic reduction at the end (ds_add_f32).
  #pragma unroll
  for (int r = 0; r < 8; ++r)
    atomicAdd(&s_score[sT * 16 + r + 8 * hi], pscore[r]);
  __syncthreads();
  if (threadIdx.x < 32)
    score[(size_t)b * SEQ + s0 + threadIdx.x] = s_score[threadIdx.x] + V_b[0];
}

// ---------------------------------------------------------------------------
// K3: softmax over S per batch. grid: 32 WGs, block 256 (8 elems/thread).
__global__ void softmax_kernel(const float* __restrict__ score,
                               float* __restrict__ weights) {
  const int b = blockIdx.x;
  __shared__ float red[256];
  float local[8];

  float mx = -3.4e38f;
  #pragma unroll
  for (int i = 0; i < 8; ++i) {
    local[i] = score[(size_t)b * SEQ + i * 256 + threadIdx.x];
    mx = fmaxf(mx, local[i]);
  }
  red[threadIdx.x] = mx; __syncthreads();
  for (int s = 128; s > 0; s >>= 1) {
    if (threadIdx.x < s) red[threadIdx.x] = fmaxf(red[threadIdx.x], red[threadIdx.x + s]);
    __syncthreads();
  }
  mx = red[0]; __syncthreads();

  float sum = 0.0f;
  #pragma unroll
  for (int i = 0; i < 8; ++i) { local[i] = __expf(local[i] - mx); sum += local[i]; }
  red[threadIdx.x] = sum; __syncthreads();
  for (int s = 128; s > 0; s >>= 1) {
    if (threadIdx.x < s) red[threadIdx.x] += red[threadIdx.x + s];
    __syncthreads();
  }
  const float inv = 1.0f / red[0];
  #pragma unroll
  for (int i = 0; i < 8; ++i)
    weights[(size_t)b * SEQ + i * 256 + threadIdx.x] = local[i] * inv;
}

// ---------------------------------------------------------------------------
// K4: context[b,u] = sum_s weights[b,s] * hidden[b,s,u]
__global__ void context_kernel(const float* __restrict__ hidden,
                               const float* __restrict__ weights,
                               float* __restrict__ context) {
  const int b = blockIdx.y;
  const int schunk = blockIdx.x;             // 256 s each
  const int u0 = threadIdx.x * 4;            // 256*4 = 1024 = UNITS
  const float* hbase = hidden + ((size_t)b * SEQ + (size_t)schunk * 256) * UNITS + u0;
  const float* wbase = weights + (size_t)b * SEQ + schunk * 256;

  v4f c = {0.f, 0.f, 0.f, 0.f};
  for (int s = 0; s < 256; ++s) {
    const float w  = wbase[s];
    const v4f   h4 = *(const v4f*)(hbase + (size_t)s * UNITS);
    c.x += w * h4.x; c.y += w * h4.y; c.z += w * h4.z; c.w += w * h4.w;
  }
  float* cp = context + (size_t)b * UNITS + u0;
  atomicAdd(cp + 0, c.x); atomicAdd(cp + 1, c.y);
  atomicAdd(cp + 2, c.z); atomicAdd(cp + 3, c.w);
}

// ---------------------------------------------------------------------------
extern "C" void kernel_launch(void* const* d_in, const int* in_sizes, int n_in,
                              void* d_out, int out_size, void* d_ws, size_t ws_size,
                              hipStream_t stream) {
  const float* s_prev = (const float*)d_in[0];
  const float* hidden = (const float*)d_in[1];
  const float* W_w    = (const float*)d_in[2];
  const float* W_b    = (const float*)d_in[3];
  const float* U_w    = (const float*)d_in[4];
  const float* U_b    = (const float*)d_in[5];
  const float* V_w    = (const float*)d_in[6];
  const float* V_b    = (const float*)d_in[7];

  float* context = (float*)d_out;                   // [B,U]   = 32768 floats
  float* weights = (float*)d_out + BATCH * UNITS;   // [B,S,1] = 65536 floats

  float* Ut    = (float*)d_ws;                      // [U,U]  1M floats (4 MB)
  float* wsb   = Ut + (size_t)UNITS * UNITS;        // [B,U]  32768 floats
  float* score = wsb + BATCH * UNITS;               // [B,S]  65536 floats

  transpose_kernel<<<dim3(32, 32), dim3(32, 8), 0, stream>>>(U_w, Ut);
  ws_gemm_kernel<<<UNITS / 64, 256, 0, stream>>>(s_prev, W_w, W_b, U_b, wsb);
  score_kernel<<<BATCH * (SEQ / 32), 256, 0, stream>>>(hidden, Ut, V_w, V_b, wsb, score);
  softmax_kernel<<<BATCH, 256, 0, stream>>>(score, weights);
  (void)hipMemsetAsync(context, 0, (size_t)BATCH * UNITS * sizeof(float), stream);
  context_kernel<<<dim3(8, BATCH), 256, 0, stream>>>(hidden, weights, context);
}